// CNN_36644660970295
// MI455X (gfx1250) — compile-verified
//
#include <hip/hip_runtime.h>
#include <hip/hip_bf16.h>
#include <stdint.h>

// Problem constants from the reference (B=32, C=1, H=W=512, steps=16, wrap pad).
#define HIMG 512
#define WIMG 512
#define BIMG 32
#define STEPS 16

// Block tile: 128 wide x 16 high, 8 waves; each wave owns a 16x16 subtile.
#define TBW 128
#define TBH 16
#define HALO_H 18          // TBH + 2
#define HALO_W 130         // TBW + 2
#define LDS_W 132          // +2 zero pad columns for the K=20 padded WMMA chunks

typedef float v2f __attribute__((ext_vector_type(2)));
typedef float v8f __attribute__((ext_vector_type(8)));

union V8u { v8f v; float f[8]; };

// ---- CDNA5 async global->LDS copy (ASYNCcnt-tracked), GV addressing mode ----
__device__ __forceinline__ void async_copy_f32(uint32_t lds_byte_off, const float* gptr) {
  // global_load_async_to_lds_b32: VDST = LDS byte offset (VGPR), VADDR = 64-bit global addr
  asm volatile("global_load_async_to_lds_b32 %0, %1, off"
               :: "v"(lds_byte_off), "v"(gptr)
               : "memory");
}

__device__ __forceinline__ void wait_async0() {
#if __has_builtin(__builtin_amdgcn_s_wait_asynccnt)
  __builtin_amdgcn_s_wait_asynccnt(0);
#else
  asm volatile("s_wait_asynccnt 0" ::: "memory");
#endif
}

// One fused step: y = relu(b2 + w2_0*tanh(conv3x3_0(h)+b1_0) + w2_1*tanh(conv3x3_1(h)+b1_1))
// with wrap padding. Conv done as banded-matrix WMMA: D_c = sum_dy A_dy x B_{c,dy}.
__global__ __launch_bounds__(256) void CNN_step_kernel(
    const float* __restrict__ src,
    const float* __restrict__ w1,   // (2,1,3,3) OIHW
    const float* __restrict__ b1,   // (2,)
    const float* __restrict__ w2,   // (1,2,1,1)
    const float* __restrict__ b2,   // (1,)
    float* __restrict__ dst)
{
  __shared__ float tile[HALO_H][LDS_W];        // halo tile of input rows
  __shared__ float bmatT[2][3][16][20];        // banded weights, transposed: [c][dy][n][k]

  const int tid = threadIdx.x;
  const int X0  = blockIdx.x * TBW;
  const int Y0  = blockIdx.y * TBH;
  const size_t ibase = (size_t)blockIdx.z * (HIMG * WIMG);

  // ---- Stage 18x130 wrap-addressed halo straight into LDS (async, no VGPR staging) ----
  const uint32_t tile_base = (uint32_t)(uintptr_t)(&tile[0][0]);
  for (int i = tid; i < HALO_H * HALO_W; i += 256) {
    const int h  = i / HALO_W;
    const int w  = i - h * HALO_W;
    const int gy = (Y0 - 1 + h) & (HIMG - 1);   // wrap
    const int gx = (X0 - 1 + w) & (WIMG - 1);   // wrap
    const float* gp = src + ibase + ((size_t)gy << 9) + gx;
    async_copy_f32(tile_base + 4u * (uint32_t)(h * LDS_W + w), gp);
  }
  // Zero the 2 pad columns (K=18,19 chunk reads land here; must be 0.0, not garbage/NaN).
  if (tid < 2 * HALO_H) {
    tile[tid >> 1][HALO_W + (tid & 1)] = 0.0f;
  }
  // ---- Build banded B matrices: B_{c,dy}(j, x) = w1[c][dy][j-x] for 0<=j-x<=2, else 0 ----
  for (int i = tid; i < 2 * 3 * 16 * 20; i += 256) {
    const int j  = i % 20;
    int t        = i / 20;
    const int x  = t % 16;  t /= 16;
    const int dy = t % 3;
    const int c  = t / 3;
    const int d  = j - x;
    (&bmatT[0][0][0][0])[i] = (d >= 0 && d <= 2) ? w1[c * 9 + dy * 3 + d] : 0.0f;
  }

  // Uniform scalars (scalar-cached loads).
  const float s_b10 = b1[0], s_b11 = b1[1];
  const float s_w20 = w2[0], s_w21 = w2[1];
  const float s_b2  = b2[0];

  wait_async0();        // ASYNCcnt -> 0 : our LDS writes landed
  __syncthreads();      // all waves' tile + bmatT visible

  // ---- Per-wave 16x16 subtile via V_WMMA_F32_16X16X4_F32 ----
  const int wave = tid >> 5;
  const int lane = tid & 31;
  const int mn   = lane & 15;   // A row (=output row) and B col (=output col) index
  const int hi   = lane >> 4;   // K-pair selector / D row-half selector
  const int xs   = wave * 16;   // subtile column origin inside block tile

  V8u acc0, acc1;
  acc0.v = (v8f){0.f, 0.f, 0.f, 0.f, 0.f, 0.f, 0.f, 0.f};
  acc1.v = (v8f){0.f, 0.f, 0.f, 0.f, 0.f, 0.f, 0.f, 0.f};

#pragma unroll
  for (int dy = 0; dy < 3; ++dy) {
#pragma unroll
    for (int kc = 0; kc < 5; ++kc) {      // K = 20 (18 real + 2 zero-padded)
      const int kb = kc * 4;
      // A(r=mn, j=kb+2*hi+{0,1}) = input row (Y0-1 + mn+dy), cols (X0-1 + xs+j)
      v2f a  = *reinterpret_cast<const v2f*>(&tile[mn + dy][xs + kb + 2 * hi]);
      v2f bw0 = *reinterpret_cast<const v2f*>(&bmatT[0][dy][mn][kb + 2 * hi]);
      v2f bw1 = *reinterpret_cast<const v2f*>(&bmatT[1][dy][mn][kb + 2 * hi]);
      acc0.v = __builtin_amdgcn_wmma_f32_16x16x4_f32(false, a, false, bw0,
                                                     (short)0, acc0.v, false, false);
      acc1.v = __builtin_amdgcn_wmma_f32_16x16x4_f32(false, a, false, bw1,
                                                     (short)0, acc1.v, false, false);
    }
  }

  // ---- Epilogue: D layout => lane holds rows j+8*hi, col mn, for BOTH channels ----
  float* dbase = dst + ibase;
#pragma unroll
  for (int j = 0; j < 8; ++j) {
    const int r = j + 8 * hi;
    const float t0 = tanhf(acc0.f[j] + s_b10);
    const float t1 = tanhf(acc1.f[j] + s_b11);
    float o = s_b2 + s_w20 * t0 + s_w21 * t1;
    o = fmaxf(o, 0.0f);
    dbase[((size_t)(Y0 + r) << 9) + (size_t)(X0 + xs + mn)] = o;
  }
}

extern "C" void kernel_launch(void* const* d_in, const int* in_sizes, int n_in,
                              void* d_out, int out_size, void* d_ws, size_t ws_size,
                              hipStream_t stream) {
  (void)in_sizes; (void)n_in; (void)out_size; (void)ws_size;
  const float* x  = (const float*)d_in[0];
  const float* w1 = (const float*)d_in[1];
  const float* b1 = (const float*)d_in[2];
  const float* w2 = (const float*)d_in[3];
  const float* b2 = (const float*)d_in[4];
  float* out = (float*)d_out;
  float* tmp = (float*)d_ws;   // needs 32 MiB ping buffer

  dim3 grid(WIMG / TBW, HIMG / TBH, BIMG);   // (4, 32, 32)
  const float* s = x;
  for (int step = 0; step < STEPS; ++step) {
    float* d = (step & 1) ? out : tmp;       // step 15 (last) lands in d_out
    CNN_step_kernel<<<grid, 256, 0, stream>>>(s, w1, b1, w2, b2, d);
    s = d;
  }
}